// CumulativeLayerNorm_34248069218360
// MI455X (gfx1250) — compile-verified
//
#include <hip/hip_runtime.h>

// Cumulative LayerNorm for MI455X (gfx1250, wave32).
// Pass 1: per-row sum / sum-of-squares via V_WMMA_F32_16X16X4_F32 (ones-vector
//         reduction), with double-buffered GLOBAL_LOAD_ASYNC_TO_LDS_B128
//         streaming (ASYNCcnt-synchronized, no block barriers) when available.
// Pass 2: per-batch cumulative scan -> mean, inv_std
// Pass 3: elementwise normalize (float4 streaming; input re-read hits 192MB L2)

typedef __attribute__((ext_vector_type(2))) float v2f;
typedef __attribute__((ext_vector_type(8))) float v8f;

#define B_ 8
#define K_ 8192
#define H_ 512
#define ROWS_PER_TILE 16
#define NTILES ((B_ * K_) / ROWS_PER_TILE)   // 4096 tiles of 16 rows
#define LDS_STRIDE 132                        // padded row stride (floats) -> conflict-free

#if defined(__has_builtin)
#if __has_builtin(__builtin_amdgcn_global_load_async_to_lds_b128) && \
    __has_builtin(__builtin_amdgcn_s_wait_asynccnt)
#define USE_ASYNC_LDS 1
#endif
#endif

#ifdef USE_ASYNC_LDS
// Builtin signature (from hipcc diagnostic): param 0/1 are pointers to a
// 16-byte int vector; clang accepts generic-AS pointers for builtin ptr args.
typedef int async_v4i __attribute__((vector_size(4 * sizeof(int))));

// Issue 16 async b128 loads: one 512B row-chunk per instruction (fully
// coalesced), written directly to this wave's LDS slice, bypassing VGPRs.
__device__ __forceinline__ void cln_issue_chunk_async(
    const float* __restrict__ gsrc, float* lbuf, int lane)
{
    #pragma unroll
    for (int r = 0; r < ROWS_PER_TILE; ++r) {
        const float* g = gsrc + (size_t)r * H_ + lane * 4;
        float*       l = lbuf + r * LDS_STRIDE + lane * 4;
        __builtin_amdgcn_global_load_async_to_lds_b128(
            (async_v4i*)g, (async_v4i*)l, /*offset=*/0, /*cpol=*/0);
    }
}
#endif

// ---------------------------------------------------------------------------
// Pass 1: for each (b,k) row of 512 floats, compute sum and sum of squares.
// One wave (32 lanes) per 16-row tile; WMMA A = 16x4 slab of x (or x*x),
// B = all-ones, C chained. D[m][0] = row sum of row m.
// ---------------------------------------------------------------------------
__global__ __launch_bounds__(128) void cln_rowstats_wmma(
    const float* __restrict__ in,
    float* __restrict__ stepSum,
    float* __restrict__ stepSq)
{
    __shared__ __align__(16) float lds[4][2][ROWS_PER_TILE][LDS_STRIDE];

    const int tid  = threadIdx.x;
    const int wave = tid >> 5;
    const int lane = tid & 31;
    const int tile = blockIdx.x * 4 + wave;
    const int row0 = tile * ROWS_PER_TILE;           // flat (b*K + k0)
    const float* src = in + (size_t)row0 * H_;

    v8f cS = {};   // accumulator for sum(x)
    v8f cQ = {};   // accumulator for sum(x*x)
    const v2f ones = {1.0f, 1.0f};

    // WMMA A-matrix layout (16x4 f32): lanes 0-15 hold M=lane, K=0..1;
    // lanes 16-31 hold M=lane-16, K=2..3. For B==ones only the row grouping matters.
    const int rowIdx = lane & 15;
    const int koff   = (lane >> 4) << 1;             // 0 or 2

#ifdef USE_ASYNC_LDS
    // ---- double-buffered async streaming: issue chunk c+1, compute chunk c.
    cln_issue_chunk_async(src, &lds[wave][0][0][0], lane);
    for (int c = 0; c < 4; ++c) {
        if (c < 3)
            cln_issue_chunk_async(src + (c + 1) * 128,
                                  &lds[wave][(c + 1) & 1][0][0], lane);
        // Async loads complete in order: <=16 outstanding => chunk c landed.
        if (c < 3) __builtin_amdgcn_s_wait_asynccnt(16);
        else       __builtin_amdgcn_s_wait_asynccnt(0);
        asm volatile("" ::: "memory");

        const float* rp = &lds[wave][c & 1][rowIdx][0];
        #pragma unroll
        for (int h4 = 0; h4 < 32; ++h4) {
            v2f a;
            a.x = rp[h4 * 4 + koff];
            a.y = rp[h4 * 4 + koff + 1];
            v2f a2 = a * a;
            cS = __builtin_amdgcn_wmma_f32_16x16x4_f32(
                     false, a,  false, ones, (short)0, cS, false, false);
            cQ = __builtin_amdgcn_wmma_f32_16x16x4_f32(
                     false, a2, false, ones, (short)0, cQ, false, false);
        }
    }
#else
    // ---- fallback: synchronous staging through VGPRs with block barriers.
    for (int c = 0; c < 4; ++c) {
        const float* s0 = src + c * 128 + lane * 4;
        #pragma unroll
        for (int r = 0; r < ROWS_PER_TILE; ++r) {
            const float4 v = *(const float4*)(s0 + (size_t)r * H_);
            *(float4*)&lds[wave][c & 1][r][lane * 4] = v;
        }
        __syncthreads();

        const float* rp = &lds[wave][c & 1][rowIdx][0];
        #pragma unroll
        for (int h4 = 0; h4 < 32; ++h4) {
            v2f a;
            a.x = rp[h4 * 4 + koff];
            a.y = rp[h4 * 4 + koff + 1];
            v2f a2 = a * a;
            cS = __builtin_amdgcn_wmma_f32_16x16x4_f32(
                     false, a,  false, ones, (short)0, cS, false, false);
            cQ = __builtin_amdgcn_wmma_f32_16x16x4_f32(
                     false, a2, false, ones, (short)0, cQ, false, false);
        }
        __syncthreads();
    }
#endif

    // C/D layout: VGPR r, lanes 0-15 -> M=r, lanes 16-31 -> M=r+8; all N equal.
    if (lane == 0 || lane == 16) {
        const int rbase = row0 + ((lane == 0) ? 0 : 8);
        #pragma unroll
        for (int r = 0; r < 8; ++r) {
            stepSum[rbase + r] = cS[r];
            stepSq [rbase + r] = cQ[r];
        }
    }
}

// ---------------------------------------------------------------------------
// Pass 2: per-batch inclusive scan of (stepSum, stepSq) over K, producing
// mean and inv_std per (b,k). One block of 256 threads per batch, 32 chunks.
// ---------------------------------------------------------------------------
__global__ __launch_bounds__(256) void cln_scan(
    const float* __restrict__ stepSum,
    const float* __restrict__ stepSq,
    float* __restrict__ meanOut,
    float* __restrict__ invOut)
{
    __shared__ float2 sc[256];
    __shared__ float2 carry;

    const int tid = threadIdx.x;
    const int b   = blockIdx.x;

    if (tid == 0) { carry.x = 0.0f; carry.y = 0.0f; }
    __syncthreads();

    for (int chunk = 0; chunk < K_ / 256; ++chunk) {
        const int i = chunk * 256 + tid;
        const int g = b * K_ + i;

        float2 v;
        v.x = stepSum[g];
        v.y = stepSq[g];
        sc[tid] = v;
        __syncthreads();

        // Hillis-Steele inclusive scan over 256 elements (pairs)
        for (int off = 1; off < 256; off <<= 1) {
            float2 t = {0.0f, 0.0f};
            if (tid >= off) t = sc[tid - off];
            __syncthreads();
            sc[tid].x += t.x;
            sc[tid].y += t.y;
            __syncthreads();
        }

        const float2 incl = sc[tid];
        const float2 c0   = carry;
        const float cs = incl.x + c0.x;
        const float cq = incl.y + c0.y;

        const float cnt  = (float)(i + 1) * (float)H_;  // exact in fp32
        const float mean = cs / cnt;
        const float var  = cq / cnt - mean * mean;
        const float inv  = rsqrtf(var + 1e-8f);

        meanOut[g] = mean;
        invOut [g] = inv;

        __syncthreads();
        if (tid == 255) { carry.x = cs; carry.y = cq; }
        __syncthreads();
    }
}

// ---------------------------------------------------------------------------
// Pass 3: out = gamma * (x - mean) * inv_std + beta, float4 streaming.
// Exact launch: one float4 per thread.
// ---------------------------------------------------------------------------
__global__ __launch_bounds__(256) void cln_apply(
    const float* __restrict__ in,
    const float* __restrict__ gamma,
    const float* __restrict__ beta,
    const float* __restrict__ meanArr,
    const float* __restrict__ invArr,
    float* __restrict__ out)
{
    const int i  = blockIdx.x * 256 + threadIdx.x;  // float4 index
    const int bk = i >> 7;                          // H/4 = 128 float4 per row
    const int h4 = i & 127;

    const float4 x  = ((const float4*)in)[i];
    const float4 g  = ((const float4*)gamma)[h4];
    const float4 be = ((const float4*)beta)[h4];
    const float  m  = meanArr[bk];
    const float  s  = invArr[bk];

    float4 o;
    o.x = g.x * (x.x - m) * s + be.x;
    o.y = g.y * (x.y - m) * s + be.y;
    o.z = g.z * (x.z - m) * s + be.z;
    o.w = g.w * (x.w - m) * s + be.w;

    ((float4*)out)[i] = o;
}

// ---------------------------------------------------------------------------
extern "C" void kernel_launch(void* const* d_in, const int* in_sizes, int n_in,
                              void* d_out, int out_size, void* d_ws, size_t ws_size,
                              hipStream_t stream)
{
    (void)in_sizes; (void)n_in; (void)out_size; (void)ws_size;

    const float* in    = (const float*)d_in[0];   // (8, 8192, 512) fp32
    const float* gamma = (const float*)d_in[1];   // (1, 512)
    const float* beta  = (const float*)d_in[2];   // (1, 512)
    float*       out   = (float*)d_out;

    // Workspace layout (floats): stepSum | stepSq | mean | inv  (4 * 65536)
    float* ws      = (float*)d_ws;
    float* stepSum = ws;
    float* stepSq  = ws + (B_ * K_);
    float* meanArr = ws + 2 * (B_ * K_);
    float* invArr  = ws + 3 * (B_ * K_);

    // Pass 1: 4096 tiles, 4 waves (tiles) per 128-thread block
    cln_rowstats_wmma<<<NTILES / 4, 128, 0, stream>>>(in, stepSum, stepSq);

    // Pass 2: one block per batch
    cln_scan<<<B_, 256, 0, stream>>>(stepSum, stepSq, meanArr, invArr);

    // Pass 3: one float4 per thread, exact grid
    const int n4 = (B_ * K_ * H_) / 4;            // 8,388,608
    cln_apply<<<n4 / 256, 256, 0, stream>>>(in, gamma, beta, meanArr, invArr, out);
}